// GraphSAGEResBlock_85899345920543
// MI455X (gfx1250) — compile-verified
//
#include <hip/hip_runtime.h>

#define DFEAT 128
#define NNODES 50000
#define NEDGES 800000
#define PB 250          // stats partial blocks
#define RPB 200         // rows per stats block  (250*200 = 50000)
#define EPSV 1e-5f
#define TILES 4         // 16-row tiles per GEMM block
#define BROWS (TILES * 16)

typedef float v2f __attribute__((ext_vector_type(2)));
typedef float v8f __attribute__((ext_vector_type(8)));

// ---------------- utility: zero scratch ----------------
__global__ void __launch_bounds__(256) k_zero(float4* __restrict__ p, int n4) {
  int i = blockIdx.x * 256 + threadIdx.x;
  if (i < n4) p[i] = make_float4(0.f, 0.f, 0.f, 0.f);
}

// ---------------- degree: one thread per edge ----------------
__global__ void __launch_bounds__(256) k_degree(const int* __restrict__ dst,
                                                float* __restrict__ deg) {
  int e = blockIdx.x * 256 + threadIdx.x;   // grid sized exactly: 800000/256
  atomicAdd(&deg[dst[e]], 1.0f);
}

// ---------------- scatter-add: one wave per edge, 4 floats/lane ----------------
__global__ void __launch_bounds__(256) k_scatter(const float* __restrict__ feat,
                                                 const int* __restrict__ src,
                                                 const int* __restrict__ dst,
                                                 float* __restrict__ agg) {
  int e    = blockIdx.x * 8 + (threadIdx.x >> 5);   // 8 waves/block, grid exact
  int lane = threadIdx.x & 31;
  int s = src[e];
  int d = dst[e];
  float4 v = ((const float4*)(feat + (size_t)s * DFEAT))[lane];
  float* o = agg + (size_t)d * DFEAT + lane * 4;
  atomicAdd(o + 0, v.x);
  atomicAdd(o + 1, v.y);
  atomicAdd(o + 2, v.z);
  atomicAdd(o + 3, v.w);
}

// ---------------- fused SAGE GEMM: out = (agg/deg)@Wl + x@Wr + bias ----------------
// [aggn | x] (64 x 256) @ [Wl ; Wr] (256 x 128) via wmma f32 16x16x4.
// 256 threads = 8 waves; wave w owns output columns [16w,16w+16) for 4 row tiles.
// Each k-step loads one B fragment and feeds 4 WMMAs (4x B-traffic amortization).
__global__ void __launch_bounds__(256)
k_sage_gemm(const float* __restrict__ agg, const float* __restrict__ deg,
            const float* __restrict__ xr,  const float* __restrict__ Wl,
            const float* __restrict__ Wr,  const float* __restrict__ bias,
            float* __restrict__ out)
{
  __shared__ float As[BROWS * 260];         // 64 rows x 256 cols, +4 pad vs banks
  const int tid  = threadIdx.x;
  const int row0 = blockIdx.x * BROWS;

  // stage A tile: 16384 floats, one float4 per thread per iteration (16 iters)
  for (int i = tid * 4; i < BROWS * 256; i += 256 * 4) {
    const int r = i >> 8;
    const int c = i & 255;
    int m = row0 + r;
    m = (m < NNODES) ? m : (NNODES - 1);    // clamp for tail block (keeps EXEC full)
    float4 v;
    if (c < DFEAT) {
      v = *(const float4*)(agg + (size_t)m * DFEAT + c);
      const float id = 1.0f / fmaxf(deg[m], 1.0f);
      v.x *= id; v.y *= id; v.z *= id; v.w *= id;
    } else {
      v = *(const float4*)(xr + (size_t)m * DFEAT + (c - DFEAT));
    }
    *(float4*)(&As[r * 260 + c]) = v;
  }
  __syncthreads();

  const int lane = tid & 31;
  const int ln   = lane & 15;
  const int hi   = lane >> 4;               // 0: K pair (k0,k0+1)  1: (k0+2,k0+3)
  const int n0   = (tid >> 5) << 4;         // wave's output column base

  v8f acc[TILES];
#pragma unroll
  for (int t = 0; t < TILES; ++t)
    acc[t] = (v8f){0.f, 0.f, 0.f, 0.f, 0.f, 0.f, 0.f, 0.f};

  const float* B = Wl;
  for (int halfk = 0; halfk < 2; ++halfk) {
    for (int k0 = 0; k0 < DFEAT; k0 += 4) {
      const int kk = halfk * DFEAT + k0 + hi * 2;
      // B 4x16 f32 layout: v0 = row k0+2*hi, v1 = row k0+2*hi+1, col n0+ln
      v2f b;
      b.x = B[(size_t)(k0 + hi * 2)     * DFEAT + n0 + ln];
      b.y = B[(size_t)(k0 + hi * 2 + 1) * DFEAT + n0 + ln];
#pragma unroll
      for (int t = 0; t < TILES; ++t) {
        // A 16x4 f32 layout: lane(ln,hi) holds row ln, K = kk (+1)
        v2f a = *(const v2f*)(&As[(t * 16 + ln) * 260 + kk]);
        acc[t] = __builtin_amdgcn_wmma_f32_16x16x4_f32(
            false, a, false, b, (short)0, acc[t], false, false);
      }
    }
    B = Wr;
  }

  // C/D layout: vgpr i, lane -> (m = i + 8*hi, n = ln)
  const int col = n0 + ln;
  const float bv = bias[col];
#pragma unroll
  for (int t = 0; t < TILES; ++t) {
#pragma unroll
    for (int i = 0; i < 8; ++i) {
      const int m = row0 + t * 16 + i + hi * 8;
      if (m < NNODES) out[(size_t)m * DFEAT + col] = acc[t][i] + bv;
    }
  }
}

// ---------------- column stats: stage 1 (per-block partial sum/sumsq) ----------------
__global__ void __launch_bounds__(256)
k_stats_partial(const float* __restrict__ h, float* __restrict__ psum,
                float* __restrict__ psq) {
  const int f    = threadIdx.x & 127;
  const int half = threadIdx.x >> 7;
  const int r0   = blockIdx.x * RPB + half;
  float s = 0.f, sq = 0.f;
  for (int i = 0; i < RPB; i += 2) {
    float v = h[(size_t)(r0 + i) * DFEAT + f];
    s  += v;
    sq += v * v;
  }
  __shared__ float sh[512];
  sh[threadIdx.x]       = s;
  sh[256 + threadIdx.x] = sq;
  __syncthreads();
  if (threadIdx.x < 128) {
    psum[blockIdx.x * DFEAT + f] = sh[f]       + sh[128 + f];
    psq [blockIdx.x * DFEAT + f] = sh[256 + f] + sh[384 + f];
  }
}

// ---------------- column stats: stage 2 (finalize scale/shift) ----------------
__global__ void __launch_bounds__(128)
k_stats_final(const float* __restrict__ psum, const float* __restrict__ psq,
              const float* __restrict__ g, const float* __restrict__ bt,
              float* __restrict__ scale, float* __restrict__ shift) {
  const int f = threadIdx.x;
  float s = 0.f, sq = 0.f;
  for (int b = 0; b < PB; ++b) {
    s  += psum[b * DFEAT + f];
    sq += psq [b * DFEAT + f];
  }
  const float inv_n = 1.0f / (float)NNODES;
  const float mu  = s * inv_n;
  const float var = sq * inv_n - mu * mu;
  const float sc  = g[f] * rsqrtf(var + EPSV);
  scale[f] = sc;
  shift[f] = bt[f] - mu * sc;
}

// ---------------- BN + ReLU apply ----------------
__global__ void __launch_bounds__(256)
k_bn_relu(const float4* __restrict__ hp, const float* __restrict__ scale,
          const float* __restrict__ shift, float4* __restrict__ out) {
  const int i = blockIdx.x * 256 + threadIdx.x;  // grid exact: 1.6M float4
  const int f = (i & 31) << 2;
  float4 v = hp[i];
  float4 r;
  r.x = fmaxf(fmaf(v.x, scale[f + 0], shift[f + 0]), 0.f);
  r.y = fmaxf(fmaf(v.y, scale[f + 1], shift[f + 1]), 0.f);
  r.z = fmaxf(fmaf(v.z, scale[f + 2], shift[f + 2]), 0.f);
  r.w = fmaxf(fmaf(v.w, scale[f + 3], shift[f + 3]), 0.f);
  out[i] = r;
}

// ---------------- BN + residual + ReLU (final) ----------------
__global__ void __launch_bounds__(256)
k_bn_res_relu(const float4* __restrict__ hp, const float* __restrict__ scale,
              const float* __restrict__ shift, const float4* __restrict__ x,
              float4* __restrict__ out) {
  const int i = blockIdx.x * 256 + threadIdx.x;
  const int f = (i & 31) << 2;
  float4 v = hp[i];
  float4 xv = x[i];
  float4 r;
  r.x = fmaxf(fmaf(v.x, scale[f + 0], shift[f + 0]) + xv.x, 0.f);
  r.y = fmaxf(fmaf(v.y, scale[f + 1], shift[f + 1]) + xv.y, 0.f);
  r.z = fmaxf(fmaf(v.z, scale[f + 2], shift[f + 2]) + xv.z, 0.f);
  r.w = fmaxf(fmaf(v.w, scale[f + 3], shift[f + 3]) + xv.w, 0.f);
  out[i] = r;
}

extern "C" void kernel_launch(void* const* d_in, const int* in_sizes, int n_in,
                              void* d_out, int out_size, void* d_ws, size_t ws_size,
                              hipStream_t stream) {
  const float* x   = (const float*)d_in[0];
  const int*   ei  = (const int*)d_in[1];
  const int*   src = ei;
  const int*   dst = ei + NEDGES;
  const float* W1l = (const float*)d_in[2];
  const float* b1  = (const float*)d_in[3];
  const float* W1r = (const float*)d_in[4];
  const float* g1  = (const float*)d_in[5];
  const float* bt1 = (const float*)d_in[6];
  const float* W2l = (const float*)d_in[7];
  const float* b2  = (const float*)d_in[8];
  const float* W2r = (const float*)d_in[9];
  const float* g2  = (const float*)d_in[10];
  const float* bt2 = (const float*)d_in[11];
  float* out = (float*)d_out;

  const size_t ND = (size_t)NNODES * DFEAT;   // 6,400,000
  float* ws    = (float*)d_ws;
  float* deg   = ws;                   // N
  float* agg   = ws + NNODES;          // N*D (contiguous after deg)
  float* hpre  = agg + ND;             // N*D
  float* h     = hpre + ND;            // N*D
  float* psum  = h + ND;               // PB*128
  float* psq   = psum + PB * DFEAT;    // PB*128
  float* scale = psq + PB * DFEAT;     // 128
  float* shift = scale + DFEAT;        // 128

  const int gemmBlocks  = (NNODES + BROWS - 1) / BROWS;  // 782 (last block tail-guarded)
  const int scatBlocks  = NEDGES / 8;    // 100000 (8 edges/block)
  const int degBlocks   = NEDGES / 256;  // 3125
  const int elemBlocks  = (int)(ND / 4 / 256);  // 6250
  const int zero1 = (int)((NNODES + ND) / 4);   // deg+agg together
  const int zero2 = (int)(ND / 4);

  // ---- layer 1 ----
  k_zero<<<(zero1 + 255) / 256, 256, 0, stream>>>((float4*)deg, zero1);
  k_degree<<<degBlocks, 256, 0, stream>>>(dst, deg);
  k_scatter<<<scatBlocks, 256, 0, stream>>>(x, src, dst, agg);
  k_sage_gemm<<<gemmBlocks, 256, 0, stream>>>(agg, deg, x, W1l, W1r, b1, hpre);
  k_stats_partial<<<PB, 256, 0, stream>>>(hpre, psum, psq);
  k_stats_final<<<1, 128, 0, stream>>>(psum, psq, g1, bt1, scale, shift);
  k_bn_relu<<<elemBlocks, 256, 0, stream>>>((const float4*)hpre, scale, shift, (float4*)h);

  // ---- layer 2 ----
  k_zero<<<(zero2 + 255) / 256, 256, 0, stream>>>((float4*)agg, zero2);
  k_scatter<<<scatBlocks, 256, 0, stream>>>(h, src, dst, agg);
  k_sage_gemm<<<gemmBlocks, 256, 0, stream>>>(agg, deg, h, W2l, W2r, b2, hpre);
  k_stats_partial<<<PB, 256, 0, stream>>>(hpre, psum, psq);
  k_stats_final<<<1, 128, 0, stream>>>(psum, psq, g2, bt2, scale, shift);
  k_bn_res_relu<<<elemBlocks, 256, 0, stream>>>((const float4*)hpre, scale, shift,
                                                (const float4*)x, (float4*)out);
}